// RelativePE_50663434224120
// MI455X (gfx1250) — compile-verified
//
#include <hip/hip_runtime.h>
#include <hip/hip_bf16.h>

// Problem constants (match reference)
#define BATCH   2
#define NHEADS  16
#define SEQLEN  2048
#define DHEAD   64
#define NREL    257          // 2*128 + 1
#define NTILES  17           // ceil(257/16)
#define NSTR    (NTILES*16)  // 272, padded LDS row stride

typedef float v2f __attribute__((ext_vector_type(2)));
typedef float v4f __attribute__((ext_vector_type(4)));
typedef float v8f __attribute__((ext_vector_type(8)));

// One workgroup = one (b,h, 16-row) tile.
//  Phase A: Qe[16 x 272] = Q_tile(16x64) * E^T(64x272) via V_WMMA_F32_16X16X4_F32 -> LDS
//  Phase B: stream 16x2048 fp32 outputs as float4 non-temporal stores, gathering
//           from the LDS Qe tile with the clamped relative index.
__global__ __launch_bounds__(256) void relpe_wmma_kernel(
    const float* __restrict__ q,    // (B,H,S,D)
    const float* __restrict__ emb,  // (257, D)
    float* __restrict__ out)        // (B,H,S,S)
{
    __shared__ float qe[16 * NSTR];   // 17408 bytes

    const int tid  = threadIdx.x;
    const int lane = tid & 31;
    const int wave = tid >> 5;        // 8 waves
    const int blk  = blockIdx.x;      // 0 .. B*H*(S/16)-1 = 4095
    const int bh   = blk >> 7;        // blk / (S/16)
    const int m0   = (blk & 127) << 4;

    const int half = lane >> 4;       // 0: lanes 0-15, 1: lanes 16-31
    const int l15  = lane & 15;

    // ---- Phase A: fp32 WMMA GEMM ----
    // A-matrix 16x4 f32 layout (ISA 7.12.2): lane L holds row M=L&15;
    // VGPR0 = K = 2*half, VGPR1 = K = 2*half+1.  Hoist all 16 K-chunks (32 VGPRs).
    const float* qrow = q + ((size_t)bh * SEQLEN + (m0 + l15)) * DHEAD + 2 * half;
    v2f a[16];
#pragma unroll
    for (int kk = 0; kk < 16; ++kk)
        a[kk] = *(const v2f*)(qrow + kk * 4);

    // Each wave owns n-tiles {wave, wave+8, wave+16<17} -> wave-uniform loop, EXEC stays full.
    for (int nt = wave; nt < NTILES; nt += 8) {
        const int n0 = nt << 4;
        int n = n0 + l15;
        if (n > NREL - 1) n = NREL - 1;     // pad columns duplicate k=256 (never read back)
        const float* erow = emb + (size_t)n * DHEAD + 2 * half;

        v8f c = {};
#pragma unroll
        for (int kk = 0; kk < 16; ++kk) {
            // B-matrix 4x16: lane L holds col N=L&15; K split across halves like A.
            v2f b = *(const v2f*)(erow + kk * 4);
            c = __builtin_amdgcn_wmma_f32_16x16x4_f32(
                    /*neg_a=*/false, a[kk], /*neg_b=*/false, b,
                    /*c_mod=*/(short)0, c, /*reuse_a=*/false, /*reuse_b=*/false);
        }

        // C/D layout: VGPR r -> row (r + 8*half), col (n0 + l15)
#pragma unroll
        for (int r = 0; r < 8; ++r)
            qe[(r + 8 * half) * NSTR + n0 + l15] = c[r];
    }

    __syncthreads();

    // ---- Phase B: bandwidth-bound scatter, float4 NT stores ----
    float* obase = out + (size_t)bh * SEQLEN * SEQLEN;
    // 16 rows * 512 float4 per row = 8192 float4 chunks; 32 per thread.
    for (int it = tid; it < 16 * (SEQLEN / 4); it += 256) {
        const int r  = it >> 9;              // local row 0..15
        const int j0 = (it & 511) << 2;      // col base
        const int ig = m0 + r;               // global row
        v4f v;
#pragma unroll
        for (int e = 0; e < 4; ++e) {
            int d = j0 + e - ig;
            int k = d < -128 ? 0 : (d > 128 ? 256 : d + 128);
            v[e] = qe[r * NSTR + k];
        }
        __builtin_nontemporal_store(v, (v4f*)(obase + (size_t)ig * SEQLEN + j0));
    }
}

extern "C" void kernel_launch(void* const* d_in, const int* in_sizes, int n_in,
                              void* d_out, int out_size, void* d_ws, size_t ws_size,
                              hipStream_t stream) {
    (void)in_sizes; (void)n_in; (void)d_ws; (void)ws_size; (void)out_size;
    const float* q   = (const float*)d_in[0];  // (2,16,2048,64) fp32
    const float* emb = (const float*)d_in[1];  // (257,64) fp32
    float* out = (float*)d_out;                // (2,16,2048,2048) fp32

    const int nblocks = BATCH * NHEADS * (SEQLEN / 16);  // 4096
    relpe_wmma_kernel<<<nblocks, 256, 0, stream>>>(q, emb, out);
}